// SCN_multi_78048145703246
// MI455X (gfx1250) — compile-verified
//
#include <hip/hip_runtime.h>
#include <hip/hip_bf16.h>
#include <stdint.h>

#define B     128
#define V     512
#define DIN   511
#define DOUT  256
#define DEPTH 8

// ---- d_out layout (floats) ----
// out        : B*DOUT                     @ 0
// h_old_stack: DEPTH*B*V*DIN              @ HOLD_OFF
// sel_stack  : DEPTH*B*DIN                @ SEL_OFF
// last_h     : B*DIN                      @ LASTH_OFF
static const long long HOLD_OFF  = (long long)B * DOUT;                       // 32768
static const long long HOLD_SZ   = (long long)DEPTH * B * V * DIN;            // 267911168
static const long long SEL_OFF   = HOLD_OFF + HOLD_SZ;                        // 267943936
static const long long LASTH_OFF = SEL_OFF + (long long)DEPTH * B * DIN;      // 268467200

// ---- workspace layout (floats) ----
static const int WS_W   = 0;                         // DEPTH*V     = 4096
static const int WS_BH  = WS_W   + DEPTH * V;        // DEPTH*DIN   = 4088
static const int WS_BF  = WS_BH  + DEPTH * DIN;      // DEPTH*DOUT  = 2048
static const int WS_IDX = WS_BF  + DEPTH * DOUT;     // DEPTH*B ints= 1024
static const int WS_F   = WS_IDX + DEPTH * B;        // DEPTH*B*DOUT= 262144
static const int WS_IW  = WS_F   + DEPTH * B * DOUT; // B*V         = 65536

typedef float v2f  __attribute__((ext_vector_type(2)));
typedef float v8f  __attribute__((ext_vector_type(8)));
typedef unsigned int u32x4 __attribute__((ext_vector_type(4)));
typedef int   i32x4 __attribute__((ext_vector_type(4)));
typedef int   i32x8 __attribute__((ext_vector_type(8)));

// -------------------------------------------------------------------------
// Kernel 0: per-depth softmax(L[i]) and base reductions w.visible_units,
// w.visible_fs (independent of scan state -> fully parallel over depth).
// -------------------------------------------------------------------------
__global__ void __launch_bounds__(512) precompute_kernel(
    const float* __restrict__ visible_units,  // V x DIN
    const float* __restrict__ L,              // DEPTH x V
    const float* __restrict__ visible_fs,     // V x DOUT
    float* __restrict__ ws_w,                 // DEPTH x V
    float* __restrict__ ws_bh,                // DEPTH x DIN
    float* __restrict__ ws_bf)                // DEPTH x DOUT
{
    __shared__ float s_red[V];
    __shared__ float s_w[V];
    const int i = blockIdx.x;
    const int tid = threadIdx.x;

    float l = L[i * V + tid];
    s_red[tid] = l;
    __syncthreads();
    for (int s = V / 2; s > 0; s >>= 1) {
        if (tid < s) s_red[tid] = fmaxf(s_red[tid], s_red[tid + s]);
        __syncthreads();
    }
    float mx = s_red[0];
    __syncthreads();
    float e = expf(l - mx);
    s_red[tid] = e;
    __syncthreads();
    for (int s = V / 2; s > 0; s >>= 1) {
        if (tid < s) s_red[tid] += s_red[tid + s];
        __syncthreads();
    }
    float w = e / s_red[0];
    s_w[tid] = w;
    ws_w[i * V + tid] = w;
    __syncthreads();

    if (tid < DIN) {
        float acc = 0.0f;
        for (int v = 0; v < V; ++v) acc = fmaf(s_w[v], visible_units[v * DIN + tid], acc);
        ws_bh[i * DIN + tid] = acc;
    }
    if (tid < DOUT) {
        float acc = 0.0f;
        for (int v = 0; v < V; ++v) acc = fmaf(s_w[v], visible_fs[v * DOUT + tid], acc);
        ws_bf[i * DOUT + tid] = acc;
    }
}

// -------------------------------------------------------------------------
// Kernel 1: the 8-step scan. One block per batch row. h/f kept implicitly
// as base + sparse corrections (<=8 modified rows tracked via idx history).
// -------------------------------------------------------------------------
__global__ void __launch_bounds__(512) scan_kernel(
    const float* __restrict__ inp,            // B x DIN
    const float* __restrict__ visible_units,  // V x DIN
    const float* __restrict__ visible_fs,     // V x DOUT
    const float* __restrict__ biases,         // DEPTH x DOUT
    const float* __restrict__ ws_w,
    const float* __restrict__ ws_bh,
    const float* __restrict__ ws_bf,
    int*   __restrict__ ws_idx,               // DEPTH x B
    float* __restrict__ ws_f,                 // DEPTH x B x DOUT
    float* __restrict__ ws_iw,                // B x V
    float* __restrict__ sel,                  // DEPTH x B x DIN  (d_out)
    float* __restrict__ lasth)                // B x DIN          (d_out)
{
    __shared__ float s_w[V];
    __shared__ float s_iw[V];
    __shared__ float red_val[V];
    __shared__ int   red_idx[V];
    __shared__ int   s_idx[DEPTH];

    const int b = blockIdx.x;
    const int tid = threadIdx.x;

    // iw0 = [1 - sum(inp[b]), inp[b]]
    red_val[tid] = (tid < DIN) ? inp[b * DIN + tid] : 0.0f;
    __syncthreads();
    for (int s = V / 2; s > 0; s >>= 1) {
        if (tid < s) red_val[tid] += red_val[tid + s];
        __syncthreads();
    }
    float total = red_val[0];
    __syncthreads();
    s_iw[tid] = (tid == 0) ? (1.0f - total) : inp[b * DIN + tid - 1];
    __syncthreads();

    for (int i = 0; i < DEPTH; ++i) {
        s_w[tid] = ws_w[i * V + tid];
        __syncthreads();

        // argmin(iw / (w + 1e-20)) with first-occurrence tie break
        float div = s_iw[tid] / (s_w[tid] + 1e-20f);
        red_val[tid] = div;
        red_idx[tid] = tid;
        __syncthreads();
        for (int s = V / 2; s > 0; s >>= 1) {
            if (tid < s) {
                float v2 = red_val[tid + s]; int i2 = red_idx[tid + s];
                float v1 = red_val[tid];     int i1 = red_idx[tid];
                if (v2 < v1 || (v2 == v1 && i2 < i1)) { red_val[tid] = v2; red_idx[tid] = i2; }
            }
            __syncthreads();
        }
        const int   idx  = red_idx[0];
        const float vals = red_val[0];
        __syncthreads();

        // iw update
        float nv = s_iw[tid] - vals * s_w[tid];
        if (tid == idx) nv = vals;
        s_iw[tid] = nv;

        // new_h = base_h + sparse corrections (latest write per row wins)
        if (tid < DIN) {
            float c = ws_bh[i * DIN + tid];
            for (int j = i - 1; j >= 0; --j) {
                int v = s_idx[j];
                bool latest = true;
                for (int j2 = j + 1; j2 < i; ++j2)
                    if (s_idx[j2] == v) latest = false;
                if (latest)
                    c = fmaf(s_w[v],
                             sel[((long long)j * B + b) * DIN + tid] - visible_units[v * DIN + tid],
                             c);
            }
            sel[((long long)i * B + b) * DIN + tid] = c;
            if (i == DEPTH - 1) lasth[b * DIN + tid] = c;
        }
        // new_f = base_f + sparse corrections + bias
        if (tid < DOUT) {
            float c = ws_bf[i * DOUT + tid];
            for (int j = i - 1; j >= 0; --j) {
                int v = s_idx[j];
                bool latest = true;
                for (int j2 = j + 1; j2 < i; ++j2)
                    if (s_idx[j2] == v) latest = false;
                if (latest)
                    c = fmaf(s_w[v],
                             ws_f[(j * B + b) * DOUT + tid] - visible_fs[v * DOUT + tid],
                             c);
            }
            c += biases[i * DOUT + tid];
            ws_f[(i * B + b) * DOUT + tid] = c;
        }
        if (tid == 0) { s_idx[i] = idx; ws_idx[i * B + b] = idx; }
        __syncthreads();
    }
    ws_iw[b * V + tid] = s_iw[tid];
}

// -------------------------------------------------------------------------
// Kernel 2: stream h_old_stack (1.07 GB) through the Tensor Data Mover.
// Each block owns a 128-row strip of one (i,b) slice; rows are staged into
// a double-buffered LDS chunk (16 rows = 32,704 B) and bulk-stored with one
// TENSOR_STORE_FROM_LDS per chunk (wave 0 issues; TENSORcnt paces reuse).
// -------------------------------------------------------------------------
__global__ void __launch_bounds__(512) hold_tdm_kernel(
    const float* __restrict__ visible_units,  // V x DIN
    const float* __restrict__ sel,            // DEPTH x B x DIN (d_out)
    const int*   __restrict__ ws_idx,         // DEPTH x B
    float* __restrict__ hold)                 // DEPTH x B x V x DIN (d_out)
{
    __shared__ float buf[2][16 * DIN];        // 2 x 32,704 B = 65,408 B

    const int ib    = blockIdx.x >> 2;        // i*B + b
    const int i     = ib >> 7;                // / B
    const int b     = ib & (B - 1);
    const int vbase = (blockIdx.x & 3) * 128; // 128-row strip
    const int tid   = threadIdx.x;

    // modification history for this (i,b): rows set by steps j < i
    int mod_idx[DEPTH];
    for (int j = 0; j < DEPTH; ++j) mod_idx[j] = (j < i) ? ws_idx[j * B + b] : -1;

    for (int c = 0; c < 8; ++c) {             // 8 chunks x 16 rows = 128 rows
        const int p = c & 1;
        if (tid < 32) __builtin_amdgcn_s_wait_tensorcnt(1);  // buffer p free
        __syncthreads();

        const int v0 = vbase + c * 16;
        for (int r = 0; r < 16; ++r) {
            const int v = v0 + r;
            const float* src = visible_units + (long long)v * DIN;
            for (int j = i - 1; j >= 0; --j)
                if (mod_idx[j] == v) { src = sel + ((long long)j * B + b) * DIN; break; }
            if (tid < DIN) buf[p][r * DIN + tid] = src[tid];
        }
        __syncthreads();

        if (tid < 32) {
            const unsigned long long gaddr = (unsigned long long)(uintptr_t)
                (hold + ((long long)ib * V + v0) * DIN);
            const unsigned int lds_off = (unsigned int)(uintptr_t)(&buf[p][0]);
            // D# group 0: count=1 | lds_addr | global_addr(57b) | type=2
            u32x4 g0;
            g0.x = 0x1u;
            g0.y = lds_off;
            g0.z = (unsigned int)(gaddr & 0xFFFFFFFFull);
            g0.w = (unsigned int)((gaddr >> 32) & 0x1FFFFFFull) | 0x80000000u;
            // D# group 1: data_size=4B, tensor_dim0=0x7FFFFFFF (no OOB drops),
            // tensor_dim1=1, tile_dim0=8176 (1-D tile), strides=8176
            i32x8 g1;
            g1[0] = 0x00020000;               // data_size=2 (4 bytes)
            g1[1] = (int)0xFFFF0000u;         // tensor_dim0[15:0] << 16
            g1[2] = 0x00017FFF;               // td0[31:16] | td1[15:0]<<16
            g1[3] = (16 * DIN) << 16;         // tile_dim0 = 8176
            g1[4] = 0;                        // tile_dim1=0, tile_dim2=0
            g1[5] = 16 * DIN;                 // tensor_dim0_stride[31:0]
            g1[6] = (16 * DIN) << 16;         // td0s[47:32]=0 | td1s[15:0]<<16
            g1[7] = 0;
            i32x4 z4 = {0, 0, 0, 0};          // groups 2/3 unused (<=2D)
            i32x8 z8 = {0, 0, 0, 0, 0, 0, 0, 0};
            __builtin_amdgcn_tensor_store_from_lds(g0, g1, z4, z4, z8, 0);
        }
    }
    if (tid < 32) __builtin_amdgcn_s_wait_tensorcnt(0);
}

// -------------------------------------------------------------------------
// Kernel 3: out = iw @ f_final. Dense part iw @ visible_fs via
// V_WMMA_F32_16X16X4_F32 (M=128,N=256,K=512), then <=8 sparse f-row
// corrections per output element. One wave per 16x16 tile, 128 tiles.
// -------------------------------------------------------------------------
__global__ void __launch_bounds__(128) out_gemm_kernel(
    const float* __restrict__ ws_iw,       // B x V (final iw)
    const float* __restrict__ visible_fs,  // V x DOUT
    const float* __restrict__ ws_f,        // DEPTH x B x DOUT (new_f rows)
    const int*   __restrict__ ws_idx,      // DEPTH x B
    float* __restrict__ out)               // B x DOUT
{
    const int lane = threadIdx.x & 31;
    const int wave = threadIdx.x >> 5;
    const int tile = blockIdx.x * 4 + wave;   // 0..127
    const int m0 = (tile >> 4) * 16;
    const int n0 = (tile & 15) * 16;
    const int ml = lane & 15;
    const int khalf = (lane >> 4) << 1;       // lanes 0-15 -> K 0/1, 16-31 -> K 2/3

    v8f acc = {};
    const float* Abase = ws_iw + (long long)(m0 + ml) * V;
    const float* Bbase = visible_fs + (n0 + ml);
    for (int k0 = 0; k0 < V; k0 += 4) {
        v2f a, bm;
        a.x  = Abase[k0 + khalf];
        a.y  = Abase[k0 + khalf + 1];
        bm.x = Bbase[(long long)(k0 + khalf) * DOUT];
        bm.y = Bbase[(long long)(k0 + khalf + 1) * DOUT];
        acc = __builtin_amdgcn_wmma_f32_16x16x4_f32(
            false, a, false, bm, (short)0, acc, false, false);
    }

    const int n  = n0 + ml;
    const int mh = (lane >> 4) << 3;
#pragma unroll
    for (int r = 0; r < 8; ++r) {
        const int m = m0 + r + mh;
        float c = acc[r];
        for (int j = DEPTH - 1; j >= 0; --j) {
            int v = ws_idx[j * B + m];
            bool latest = true;
            for (int j2 = j + 1; j2 < DEPTH; ++j2)
                if (ws_idx[j2 * B + m] == v) latest = false;
            if (latest)
                c = fmaf(ws_iw[m * V + v],
                         ws_f[(j * B + m) * DOUT + n] - visible_fs[v * DOUT + n],
                         c);
        }
        out[m * DOUT + n] = c;
    }
}

// -------------------------------------------------------------------------
extern "C" void kernel_launch(void* const* d_in, const int* in_sizes, int n_in,
                              void* d_out, int out_size, void* d_ws, size_t ws_size,
                              hipStream_t stream) {
    const float* inp           = (const float*)d_in[0];  // B x DIN
    const float* visible_units = (const float*)d_in[1];  // V x DIN
    const float* L             = (const float*)d_in[2];  // DEPTH x V
    const float* visible_fs    = (const float*)d_in[3];  // V x DOUT
    const float* biases        = (const float*)d_in[4];  // DEPTH x DOUT

    float* out   = (float*)d_out;
    float* hold  = out + HOLD_OFF;
    float* sel   = out + SEL_OFF;
    float* lasth = out + LASTH_OFF;

    float* ws     = (float*)d_ws;
    float* ws_w   = ws + WS_W;
    float* ws_bh  = ws + WS_BH;
    float* ws_bf  = ws + WS_BF;
    int*   ws_idx = (int*)(ws + WS_IDX);
    float* ws_f   = ws + WS_F;
    float* ws_iw  = ws + WS_IW;

    precompute_kernel<<<DEPTH, 512, 0, stream>>>(visible_units, L, visible_fs,
                                                 ws_w, ws_bh, ws_bf);
    scan_kernel<<<B, 512, 0, stream>>>(inp, visible_units, visible_fs, biases,
                                       ws_w, ws_bh, ws_bf, ws_idx, ws_f, ws_iw,
                                       sel, lasth);
    hold_tdm_kernel<<<DEPTH * B * 4, 512, 0, stream>>>(visible_units, sel, ws_idx, hold);
    out_gemm_kernel<<<32, 128, 0, stream>>>(ws_iw, visible_fs, ws_f, ws_idx, out);
}